// ReverseGenerator_75067438399601
// MI455X (gfx1250) — compile-verified
//
#include <hip/hip_runtime.h>
#include <hip/hip_bf16.h>
#include <stdint.h>

#define DMODEL 64
#define NHEADS 4
#define DHEAD 16
#define FFDIM 2048
#define VOCAB 1000000
#define STEPS 10
#define LOGITS_WG 1024
#define LN_EPS 1e-5f

typedef float v2f __attribute__((ext_vector_type(2)));
typedef float v8f __attribute__((ext_vector_type(8)));

// ---------------------------------------------------------------------------
// Wave-level fp32 WMMA step: C(16x16) += A(16x4) * B(4x16)
// A layout: lanes 0-15 hold M=0..15 (K pair {k0,k0+1}), lanes 16-31 K pair {k0+2,k0+3}
// B layout: lanes hold N=lane&15; VGPR0/1 hold K rows per half-wave (mirrors A)
// C layout: VGPR r -> row (r + 8*(lane>=16)), col lane&15
// ---------------------------------------------------------------------------
__device__ __forceinline__ v8f wmma_k4(v8f c, float ax, float ay, float bx, float by) {
    v2f a, b;
    a.x = ax; a.y = ay;
    b.x = bx; b.y = by;
    return __builtin_amdgcn_wmma_f32_16x16x4_f32(false, a, false, b, (short)0, c, false, false);
}

// ---------------------------------------------------------------------------
// K0: item_emb fp32 -> bf16 (RNE), packed 2 per dword.  n = VOCAB*DMODEL/2
// ---------------------------------------------------------------------------
__global__ __launch_bounds__(256) void k_cvt(const float* __restrict__ src,
                                             unsigned int* __restrict__ dst, int n) {
    int i = blockIdx.x * 256 + threadIdx.x;
    if (i >= n) return;
    float2 f = ((const float2*)src)[i];
    unsigned ux = __float_as_uint(f.x); ux = ux + 0x7FFFu + ((ux >> 16) & 1u);
    unsigned uy = __float_as_uint(f.y); uy = uy + 0x7FFFu + ((uy >> 16) & 1u);
    dst[i] = (ux >> 16) | (uy & 0xFFFF0000u);
}

// ---------------------------------------------------------------------------
// K1: build x (step s, layer l), QKV proj (WMMA), attention, out-proj (WMMA),
//     residual + LN1 -> X1 (16x64, rows >= S zeroed)
// ---------------------------------------------------------------------------
__global__ __launch_bounds__(256) void k_attn(
    int s, int l,
    const float* __restrict__ ht, const float* __restrict__ item_emb,
    const float* __restrict__ pos_emb, const int* __restrict__ gen,
    const float* __restrict__ ipw, const float* __restrict__ ipb,
    const float* __restrict__ ow, const float* __restrict__ obias,
    const float* __restrict__ l1w, const float* __restrict__ l1b,
    const float* __restrict__ Xin, float* __restrict__ X1out) {
    const int S = s + 1;
    __shared__ float xb[16][64];      // input activations (padded to 16 rows)
    __shared__ float qk[16][192];     // qkv, later reused for y = x + attn_out
    __shared__ float oatt[16][64];    // attention output
    const int t = threadIdx.x;
    const int lane = t & 31, wave = t >> 5;
    const int nm = lane & 15;                 // serves as M for A and N for B
    const int hi = (lane >> 4) << 1;          // K sub-offset {0,2}
    const int rbase = (lane >> 4) << 3;       // C row base {0,8}

    // ---- build x = stack + pos_emb (layer 0) or load Xin (layer 1) ----
    for (int i = t; i < 16 * 64; i += 256) {
        int r = i >> 6, d = i & 63;
        float v = 0.f;
        if (r < S) {
            if (l == 0) {
                float base = (r == 0) ? ht[d]
                                      : item_emb[(size_t)gen[r - 1] * DMODEL + d];
                v = base + pos_emb[r * DMODEL + d];
            } else {
                v = Xin[r * DMODEL + d];
            }
        }
        xb[r][d] = v;
    }
    __syncthreads();

    // ---- qkv = x @ ipw.T + ipb : 12 tiles of 16 cols, K=64 ----
    for (int tile = wave; tile < 12; tile += 8) {
        v8f acc = {};
        const float* bb = ipw + (size_t)(tile * 16 + nm) * 64;  // B[k][n] = ipw[n][k]
        #pragma unroll
        for (int k0 = 0; k0 < 64; k0 += 4)
            acc = wmma_k4(acc, xb[nm][k0 + hi], xb[nm][k0 + hi + 1],
                          bb[k0 + hi], bb[k0 + hi + 1]);
        float bias = ipb[tile * 16 + nm];
        #pragma unroll
        for (int r = 0; r < 8; ++r)
            qk[rbase + r][tile * 16 + nm] = acc[r] + bias;
    }
    __syncthreads();

    // ---- attention: thread (h, sq), 4*S threads ----
    if (t < NHEADS * S) {
        int h = t / S, sq = t - h * S;
        float q[DHEAD], sc[STEPS];
        #pragma unroll
        for (int d = 0; d < DHEAD; ++d) q[d] = qk[sq][h * DHEAD + d];
        float mx = -1e30f;
        for (int tk = 0; tk < S; ++tk) {
            float a = 0.f;
            #pragma unroll
            for (int d = 0; d < DHEAD; ++d) a += q[d] * qk[tk][64 + h * DHEAD + d];
            a *= 0.25f;  // 1/sqrt(16)
            sc[tk] = a; mx = fmaxf(mx, a);
        }
        float sum = 0.f;
        for (int tk = 0; tk < S; ++tk) { sc[tk] = __expf(sc[tk] - mx); sum += sc[tk]; }
        float inv = 1.f / sum;
        #pragma unroll
        for (int d = 0; d < DHEAD; ++d) {
            float o = 0.f;
            for (int tk = 0; tk < S; ++tk) o += sc[tk] * qk[tk][128 + h * DHEAD + d];
            oatt[sq][h * DHEAD + d] = o * inv;
        }
    }
    // zero padded rows of attention output
    for (int i = t; i < 16 * 64; i += 256) {
        int r = i >> 6;
        if (r >= S) oatt[r][i & 63] = 0.f;
    }
    __syncthreads();

    // ---- out proj (4 tiles) + residual into qk[.][0..63] ----
    if (wave < 4) {
        int tile = wave;
        v8f acc = {};
        const float* bb = ow + (size_t)(tile * 16 + nm) * 64;  // B[k][n] = ow[n][k]
        #pragma unroll
        for (int k0 = 0; k0 < 64; k0 += 4)
            acc = wmma_k4(acc, oatt[nm][k0 + hi], oatt[nm][k0 + hi + 1],
                          bb[k0 + hi], bb[k0 + hi + 1]);
        float bias = obias[tile * 16 + nm];
        #pragma unroll
        for (int r = 0; r < 8; ++r) {
            int row = rbase + r;
            qk[row][tile * 16 + nm] = xb[row][tile * 16 + nm] + acc[r] + bias;
        }
    }
    __syncthreads();

    // ---- LN1 per row, write X1 (rows >= S zeroed) ----
    if (t < 16) {
        int r = t;
        if (r < S) {
            float m = 0.f;
            for (int d = 0; d < 64; ++d) m += qk[r][d];
            m *= (1.f / 64.f);
            float var = 0.f;
            for (int d = 0; d < 64; ++d) { float df = qk[r][d] - m; var += df * df; }
            var *= (1.f / 64.f);
            float rs = rsqrtf(var + LN_EPS);
            for (int d = 0; d < 64; ++d)
                X1out[r * 64 + d] = (qk[r][d] - m) * rs * l1w[d] + l1b[d];
        } else {
            for (int d = 0; d < 64; ++d) X1out[r * 64 + d] = 0.f;
        }
    }
}

// ---------------------------------------------------------------------------
// K2: H = relu(X1 @ f1w.T + f1b), 16 blocks x 8 waves = 128 tiles of 16 cols
// ---------------------------------------------------------------------------
__global__ __launch_bounds__(256) void k_ffn1(const float* __restrict__ X1,
                                              const float* __restrict__ f1w,
                                              const float* __restrict__ f1b,
                                              float* __restrict__ H) {
    __shared__ float xb[16][64];
    const int t = threadIdx.x;
    for (int i = t; i < 1024; i += 256) xb[i >> 6][i & 63] = X1[i];
    __syncthreads();
    const int lane = t & 31, wave = t >> 5;
    const int nm = lane & 15, hi = (lane >> 4) << 1, rbase = (lane >> 4) << 3;
    const int tile = blockIdx.x * 8 + wave;
    v8f acc = {};
    const float* bb = f1w + (size_t)(tile * 16 + nm) * 64;  // B[k][n] = f1w[n][k]
    #pragma unroll
    for (int k0 = 0; k0 < 64; k0 += 4)
        acc = wmma_k4(acc, xb[nm][k0 + hi], xb[nm][k0 + hi + 1],
                      bb[k0 + hi], bb[k0 + hi + 1]);
    float bias = f1b[tile * 16 + nm];
    #pragma unroll
    for (int r = 0; r < 8; ++r)
        H[(size_t)(rbase + r) * FFDIM + tile * 16 + nm] = fmaxf(acc[r] + bias, 0.f);
}

// ---------------------------------------------------------------------------
// K3: Y = H @ f2w.T + f2b ; x = LN2(X1 + Y).  8 waves: 4 N-tiles x 2 K-halves,
//     LDS combine, then LN + residual.  Layer 1 also emits x_last.
// ---------------------------------------------------------------------------
__global__ __launch_bounds__(256) void k_ffn2(
    int s, int l, const float* __restrict__ H, const float* __restrict__ f2w,
    const float* __restrict__ f2b, const float* __restrict__ X1,
    const float* __restrict__ l2w, const float* __restrict__ l2b,
    float* __restrict__ Xout, float* __restrict__ xlast) {
    __shared__ float part[8][16][16];
    __shared__ float yb[16][64];
    const int t = threadIdx.x;
    const int lane = t & 31, wave = t >> 5;
    const int nm = lane & 15, hi = (lane >> 4) << 1, rbase = (lane >> 4) << 3;
    const int tile = wave & 3, kh = wave >> 2;  // K half: kh*1024

    v8f acc = {};
    const float* ab = H + (size_t)nm * FFDIM + kh * 1024;                 // A[m][k]
    const float* bb = f2w + (size_t)(tile * 16 + nm) * FFDIM + kh * 1024; // B[k][n]=f2w[n][k]
    for (int k0 = 0; k0 < 1024; k0 += 4)
        acc = wmma_k4(acc, ab[k0 + hi], ab[k0 + hi + 1], bb[k0 + hi], bb[k0 + hi + 1]);
    #pragma unroll
    for (int r = 0; r < 8; ++r) part[wave][rbase + r][nm] = acc[r];
    __syncthreads();

    // combine K halves + bias + residual into yb
    for (int i = t; i < 16 * 64; i += 256) {
        int r = i >> 6, n = i & 63;
        int tl = n >> 4, c = n & 15;
        yb[r][n] = part[tl][r][c] + part[tl + 4][r][c] + f2b[n] + X1[r * 64 + n];
    }
    __syncthreads();

    const int S = s + 1;
    if (t < 16) {
        int r = t;
        if (r < S) {
            float m = 0.f;
            for (int n = 0; n < 64; ++n) m += yb[r][n];
            m *= (1.f / 64.f);
            float var = 0.f;
            for (int n = 0; n < 64; ++n) { float d = yb[r][n] - m; var += d * d; }
            var *= (1.f / 64.f);
            float rs = rsqrtf(var + LN_EPS);
            for (int n = 0; n < 64; ++n) {
                float o = (yb[r][n] - m) * rs * l2w[n] + l2b[n];
                Xout[r * 64 + n] = o;
                if (l == 1 && r == S - 1) xlast[n] = o;
            }
        } else {
            for (int n = 0; n < 64; ++n) Xout[r * 64 + n] = 0.f;
        }
    }
}

// ---------------------------------------------------------------------------
// K4: logits[v] = dot(x_last, emb[v]) + per-workgroup argmax candidate.
// BF16 path streams the L2-resident 128MB table; fp32 path is the fallback.
// ---------------------------------------------------------------------------
__device__ __forceinline__ unsigned ordkey(float f) {
    unsigned u = __float_as_uint(f);
    return (u & 0x80000000u) ? ~u : (u | 0x80000000u);
}

template <bool BF16>
__global__ __launch_bounds__(256) void k_logits(const void* __restrict__ emb,
                                                const float* __restrict__ xlast,
                                                float* __restrict__ logits,
                                                unsigned long long* __restrict__ wgbest) {
    __shared__ float xl[64];
    __shared__ unsigned long long sbest[8];
    const int t = threadIdx.x;
    if (t < 64) xl[t] = xlast[t];
    __syncthreads();

    unsigned long long best = 0ull;
    const int total = gridDim.x * 256;
    for (int v = blockIdx.x * 256 + t; v < VOCAB; v += total) {
        float acc = 0.f;
        if (BF16) {
            const uint4* p = (const uint4*)emb + (size_t)v * 8;  // 128B row
            #pragma unroll
            for (int i = 0; i < 8; ++i) {
                uint4 w = p[i];
                acc = fmaf(__uint_as_float(w.x << 16),        xl[8 * i + 0], acc);
                acc = fmaf(__uint_as_float(w.x & 0xFFFF0000u), xl[8 * i + 1], acc);
                acc = fmaf(__uint_as_float(w.y << 16),        xl[8 * i + 2], acc);
                acc = fmaf(__uint_as_float(w.y & 0xFFFF0000u), xl[8 * i + 3], acc);
                acc = fmaf(__uint_as_float(w.z << 16),        xl[8 * i + 4], acc);
                acc = fmaf(__uint_as_float(w.z & 0xFFFF0000u), xl[8 * i + 5], acc);
                acc = fmaf(__uint_as_float(w.w << 16),        xl[8 * i + 6], acc);
                acc = fmaf(__uint_as_float(w.w & 0xFFFF0000u), xl[8 * i + 7], acc);
            }
        } else {
            const float4* p = (const float4*)emb + (size_t)v * 16;  // 256B row
            #pragma unroll
            for (int i = 0; i < 16; ++i) {
                float4 w = p[i];
                acc = fmaf(w.x, xl[4 * i + 0], acc);
                acc = fmaf(w.y, xl[4 * i + 1], acc);
                acc = fmaf(w.z, xl[4 * i + 2], acc);
                acc = fmaf(w.w, xl[4 * i + 3], acc);
            }
        }
        logits[v] = acc;
        // pack (ordered_float, ~index) so max -> (max logit, lowest index on tie)
        unsigned long long pk =
            ((unsigned long long)ordkey(acc) << 32) | (0xFFFFFFFFu - (unsigned)v);
        best = pk > best ? pk : best;
    }
    // wave32 reduction
    #pragma unroll
    for (int off = 16; off; off >>= 1) {
        unsigned long long o = __shfl_xor(best, off, 32);
        best = o > best ? o : best;
    }
    if ((t & 31) == 0) sbest[t >> 5] = best;
    __syncthreads();
    if (t == 0) {
        unsigned long long b = sbest[0];
        #pragma unroll
        for (int i = 1; i < 8; ++i) b = sbest[i] > b ? sbest[i] : b;
        wgbest[blockIdx.x] = b;
    }
}

// ---------------------------------------------------------------------------
// K5: reduce per-WG candidates -> gen[s] (int to ws, float to d_out tail)
// ---------------------------------------------------------------------------
__global__ __launch_bounds__(256) void k_argmax(int s,
                                                const unsigned long long* __restrict__ wgbest,
                                                int nwg, int* __restrict__ gen,
                                                float* __restrict__ out_gen) {
    __shared__ unsigned long long sb[256];
    const int t = threadIdx.x;
    unsigned long long b = 0ull;
    for (int i = t; i < nwg; i += 256) b = wgbest[i] > b ? wgbest[i] : b;
    sb[t] = b;
    __syncthreads();
    for (int off = 128; off; off >>= 1) {
        if (t < off) sb[t] = sb[t + off] > sb[t] ? sb[t + off] : sb[t];
        __syncthreads();
    }
    if (t == 0) {
        unsigned idx = 0xFFFFFFFFu - (unsigned)(sb[0] & 0xFFFFFFFFu);
        gen[s] = (int)idx;
        out_gen[s] = (float)idx;
    }
}

// ---------------------------------------------------------------------------
extern "C" void kernel_launch(void* const* d_in, const int* in_sizes, int n_in,
                              void* d_out, int out_size, void* d_ws, size_t ws_size,
                              hipStream_t stream) {
    const float* ht       = (const float*)d_in[0];
    const float* item_emb = (const float*)d_in[1];
    const float* pos_emb  = (const float*)d_in[2];
    const float* ipw      = (const float*)d_in[3];
    const float* ipb      = (const float*)d_in[4];
    const float* ow       = (const float*)d_in[5];
    const float* obias    = (const float*)d_in[6];
    const float* l1w      = (const float*)d_in[7];
    const float* l1b      = (const float*)d_in[8];
    const float* l2w      = (const float*)d_in[9];
    const float* l2b      = (const float*)d_in[10];
    const float* f1w      = (const float*)d_in[11];
    const float* f1b      = (const float*)d_in[12];
    const float* f2w      = (const float*)d_in[13];
    const float* f2b      = (const float*)d_in[14];
    float* out = (float*)d_out;
    char* ws = (char*)d_ws;

    const size_t bfBytes = (size_t)VOCAB * DMODEL * 2;  // 128 MB bf16 table
    const bool use_bf16 = ws_size >= bfBytes + (1u << 20);

    size_t off = use_bf16 ? ((bfBytes + 255) & ~(size_t)255) : 0;
    float* X     = (float*)(ws + off); off += 16 * 64 * 4;
    float* X1    = (float*)(ws + off); off += 16 * 64 * 4;
    float* H     = (float*)(ws + off); off += (size_t)16 * FFDIM * 4;
    float* xlast = (float*)(ws + off); off += 256;
    int*   gen   = (int*)(ws + off);   off += 256;
    unsigned long long* wgbest = (unsigned long long*)(ws + off);

    if (use_bf16) {
        const int npair = VOCAB * DMODEL / 2;  // 32,000,000
        k_cvt<<<(npair + 255) / 256, 256, 0, stream>>>(item_emb, (unsigned int*)ws, npair);
    }

    for (int s = 0; s < STEPS; ++s) {
        for (int l = 0; l < 2; ++l) {
            k_attn<<<1, 256, 0, stream>>>(
                s, l, ht, item_emb, pos_emb, gen,
                ipw + (size_t)l * 3 * DMODEL * DMODEL, ipb + (size_t)l * 3 * DMODEL,
                ow + (size_t)l * DMODEL * DMODEL, obias + (size_t)l * DMODEL,
                l1w + (size_t)l * DMODEL, l1b + (size_t)l * DMODEL, X, X1);
            k_ffn1<<<16, 256, 0, stream>>>(
                X1, f1w + (size_t)l * FFDIM * DMODEL, f1b + (size_t)l * FFDIM, H);
            k_ffn2<<<1, 256, 0, stream>>>(
                s, l, H, f2w + (size_t)l * DMODEL * FFDIM, f2b + (size_t)l * DMODEL,
                X1, l2w + (size_t)l * DMODEL, l2b + (size_t)l * DMODEL, X, xlast);
        }
        if (use_bf16)
            k_logits<true><<<LOGITS_WG, 256, 0, stream>>>(
                (const void*)ws, xlast, out + (size_t)s * VOCAB, wgbest);
        else
            k_logits<false><<<LOGITS_WG, 256, 0, stream>>>(
                (const void*)item_emb, xlast, out + (size_t)s * VOCAB, wgbest);
        k_argmax<<<1, 256, 0, stream>>>(s, wgbest, LOGITS_WG, gen,
                                        out + (size_t)STEPS * VOCAB);
    }
}